// ProbabilityGraph_38482906972426
// MI455X (gfx1250) — compile-verified
//
#include <hip/hip_runtime.h>
#include <cstdint>

#define N_NODES 50000
#define N_EDGES 800000
#define D_FEAT  128

typedef float v2f __attribute__((ext_vector_type(2)));
typedef float v4f __attribute__((ext_vector_type(4)));
typedef float v8f __attribute__((ext_vector_type(8)));

// ---------------------------------------------------------------------------
// Cross-lane (wave32) reductions.
// ---------------------------------------------------------------------------
__device__ __forceinline__ float wave_max(float x) {
#pragma unroll
  for (int off = 16; off >= 1; off >>= 1)
    x = fmaxf(x, __shfl_xor(x, off, 32));
  return x;
}

// Sum of one float per lane across all 32 lanes, result broadcast to every
// lane. Uses V_WMMA_F32_16X16X4_F32 (exact f32):
//   A(16x4): vgpr0 = x (lanes 0-15 -> K0 rows 0-15, lanes 16-31 -> K2),
//            vgpr1 = 0 (K1/K3).  B(4x16) = all ones.
//   => D[m][n] = x[m] + x[m+16]  (independent of n).
// D vgpr j carries row m=j (lanes 0-15) and m=j+8 (lanes 16-31), so summing
// the 8 D vgprs gives half-sums per lane-half; one xor-16 shuffle finishes.
__device__ __forceinline__ float wave_sum_wmma(float x) {
  v2f a; a[0] = x;    a[1] = 0.0f;
  v2f b; b[0] = 1.0f; b[1] = 1.0f;
  v8f c = {};
  v8f d = __builtin_amdgcn_wmma_f32_16x16x4_f32(
      /*neg_a=*/false, a, /*neg_b=*/false, b,
      /*c_mod=*/(short)0, c, /*reuse_a=*/false, /*reuse_b=*/false);
  float t = ((d[0] + d[1]) + (d[2] + d[3])) + ((d[4] + d[5]) + (d[6] + d[7]));
  t += __shfl_xor(t, 16, 32);
  return t;
}

// ---------------------------------------------------------------------------
// Kernel 1: agg[dst] += edge_att * node_att[src]   (one wave per edge,
// float4 per lane -> 512B coalesced row accesses; native f32 atomics into
// the 25.6MB L2-resident accumulator).
// ---------------------------------------------------------------------------
__global__ void pg_scatter(const float* __restrict__ node_att,
                           const float* __restrict__ edge_att,
                           const long long* __restrict__ src,
                           const long long* __restrict__ dst,
                           float* __restrict__ agg) {
  int e = blockIdx.x * (blockDim.x >> 5) + (threadIdx.x >> 5);
  if (e >= N_EDGES) return;
  int lane = threadIdx.x & 31;
  int f = lane << 2;

  long long s = src[e];
  long long d = dst[e];

  v4f ea = *(const v4f*)(edge_att + (size_t)e * D_FEAT + f);
  v4f na = *(const v4f*)(node_att + (size_t)s * D_FEAT + f);
  v4f m  = ea * na;

  float* ap = agg + (size_t)d * D_FEAT + f;
  unsafeAtomicAdd(ap + 0, m[0]);
  unsafeAtomicAdd(ap + 1, m[1]);
  unsafeAtomicAdd(ap + 2, m[2]);
  unsafeAtomicAdd(ap + 3, m[3]);
}

// ---------------------------------------------------------------------------
// Kernel 2: new_node = softmax(agg + node_att, axis=1)  (one wave per row)
// ---------------------------------------------------------------------------
__global__ void pg_node_softmax(const float* __restrict__ node_att,
                                const float* __restrict__ agg,
                                float* __restrict__ out_node) {
  int n = blockIdx.x * (blockDim.x >> 5) + (threadIdx.x >> 5);
  if (n >= N_NODES) return;
  int lane = threadIdx.x & 31;
  size_t base = (size_t)n * D_FEAT + (lane << 2);

  v4f a = *(const v4f*)(agg + base);
  v4f b = *(const v4f*)(node_att + base);
  v4f x = a + b;

  float m = wave_max(fmaxf(fmaxf(x[0], x[1]), fmaxf(x[2], x[3])));

  v4f ex;
  ex[0] = __expf(x[0] - m);
  ex[1] = __expf(x[1] - m);
  ex[2] = __expf(x[2] - m);
  ex[3] = __expf(x[3] - m);

  float s = wave_sum_wmma((ex[0] + ex[1]) + (ex[2] + ex[3]));
  float inv = 1.0f / s;

  v4f r = ex * inv;
  *(v4f*)(out_node + base) = r;
}

// ---------------------------------------------------------------------------
// Kernel 3: new_edge = softmax(new_node[src]*new_node[dst] + edge_att, axis=1)
// (one wave per edge)
// ---------------------------------------------------------------------------
__global__ void pg_edge_softmax(const float* __restrict__ new_node,
                                const float* __restrict__ edge_att,
                                const long long* __restrict__ src,
                                const long long* __restrict__ dst,
                                float* __restrict__ out_edge) {
  int e = blockIdx.x * (blockDim.x >> 5) + (threadIdx.x >> 5);
  if (e >= N_EDGES) return;
  int lane = threadIdx.x & 31;
  int f = lane << 2;

  long long s = src[e];
  long long d = dst[e];

  v4f ns = *(const v4f*)(new_node + (size_t)s * D_FEAT + f);
  v4f nd = *(const v4f*)(new_node + (size_t)d * D_FEAT + f);
  v4f ea = *(const v4f*)(edge_att + (size_t)e * D_FEAT + f);
  v4f x  = ns * nd + ea;

  float m = wave_max(fmaxf(fmaxf(x[0], x[1]), fmaxf(x[2], x[3])));

  v4f ex;
  ex[0] = __expf(x[0] - m);
  ex[1] = __expf(x[1] - m);
  ex[2] = __expf(x[2] - m);
  ex[3] = __expf(x[3] - m);

  float ssum = wave_sum_wmma((ex[0] + ex[1]) + (ex[2] + ex[3]));
  float inv = 1.0f / ssum;

  v4f r = ex * inv;
  *(v4f*)(out_edge + (size_t)e * D_FEAT + f) = r;
}

// ---------------------------------------------------------------------------
// Launch: memset(agg) -> scatter -> node softmax -> edge softmax
// d_out layout: [N_NODES*D_FEAT floats new_node | N_EDGES*D_FEAT floats new_edge]
// d_ws: agg accumulator, N_NODES*D_FEAT floats (25.6 MB), re-zeroed per call.
// ---------------------------------------------------------------------------
extern "C" void kernel_launch(void* const* d_in, const int* in_sizes, int n_in,
                              void* d_out, int out_size, void* d_ws, size_t ws_size,
                              hipStream_t stream) {
  const float*     node_att = (const float*)d_in[0];
  const float*     edge_att = (const float*)d_in[1];
  const long long* src      = (const long long*)d_in[2];   // int64 per reference
  const long long* dst      = (const long long*)d_in[3];

  float* out_node = (float*)d_out;
  float* out_edge = out_node + (size_t)N_NODES * D_FEAT;
  float* agg      = (float*)d_ws;

  hipMemsetAsync(agg, 0, (size_t)N_NODES * D_FEAT * sizeof(float), stream);

  const int threads = 256;            // 8 wave32 per block, one row per wave
  const int rows_per_block = threads / 32;

  int eg = (N_EDGES + rows_per_block - 1) / rows_per_block;  // 100000 exact
  int ng = (N_NODES + rows_per_block - 1) / rows_per_block;  // 6250 exact

  pg_scatter<<<eg, threads, 0, stream>>>(node_att, edge_att, src, dst, agg);
  pg_node_softmax<<<ng, threads, 0, stream>>>(node_att, agg, out_node);
  pg_edge_softmax<<<eg, threads, 0, stream>>>(out_node, edge_att, src, dst, out_edge);
}